// ComputeTargets_81149112090780
// MI455X (gfx1250) — compile-verified
//
#include <hip/hip_runtime.h>
#include <cstdint>

#define NUM_CLASSES 80
#define NB 8         // batch
#define NG 128       // GT boxes per image
#define NA 131072    // anchors

// Low 32 bits of a generic pointer to __shared__ = LDS byte offset.
__device__ __forceinline__ unsigned lds_off_u32(const void* p) {
  return (unsigned)(unsigned long long)p;
}

// gfx1250 async tensor path: global -> LDS, 16B per lane, tracked by ASYNCcnt.
__device__ __forceinline__ void async_load_b128_to_lds(unsigned lds_off, const void* gaddr) {
  asm volatile("global_load_async_to_lds_b128 %0, %1, off"
               :: "v"(lds_off), "v"(gaddr)
               : "memory");
}

__device__ __forceinline__ void wait_asynccnt0() {
  asm volatile("s_wait_asynccnt 0x0" ::: "memory");
}

__global__ __launch_bounds__(256) void anchor_assign_kernel(
    const float* __restrict__ ann,   // (B,G,5)
    const float* __restrict__ anc,   // (A,4)
    float* __restrict__ cls,         // (B,A,80)
    float* __restrict__ reg,         // (B,A,4)
    float* __restrict__ st,          // (B,A)
    int*   __restrict__ codes,       // (B,A) workspace, may be null
    int    inline_cls)
{
  __shared__ float  sraw[NG * 5];    // raw staged annotations (2560 B)
  __shared__ float4 sbox[NG];        // 16B-aligned box coords
  __shared__ float  sarea[NG];       // precomputed box areas
  __shared__ int    slab[NG];        // integer labels

  const int tid = threadIdx.x;
  const int blocksPerImg = NA / 256;                 // 512
  const int b   = blockIdx.x / blocksPerImg;
  const int a   = (blockIdx.x % blocksPerImg) * 256 + tid;
  const int gid = b * NA + a;

  // ---- Stage this image's annotations into LDS via async 128-bit loads ----
  if (tid < (NG * 5) / 4) {          // 160 lanes x 16 B = 2560 B
    const float* gp = ann + (size_t)b * NG * 5 + (size_t)tid * 4;
    async_load_b128_to_lds(lds_off_u32(&sraw[tid * 4]), gp);
  }
  wait_asynccnt0();
  __syncthreads();

  // ---- Reshape to aligned SoA in LDS; precompute areas & labels ----
  if (tid < NG) {
    float x1 = sraw[tid * 5 + 0];
    float y1 = sraw[tid * 5 + 1];
    float x2 = sraw[tid * 5 + 2];
    float y2 = sraw[tid * 5 + 3];
    sbox[tid]  = make_float4(x1, y1, x2, y2);
    sarea[tid] = (x2 - x1) * (y2 - y1);
    slab[tid]  = (int)sraw[tid * 5 + 4];
  }
  __syncthreads();

  // ---- Per-anchor IoU argmax over G boxes (division-free) ----
  const float4 av = *(const float4*)(anc + (size_t)a * 4);
  const float ax1 = av.x, ay1 = av.y, ax2 = av.z, ay2 = av.w;
  const float areaA = (ax2 - ax1) * (ay2 - ay1);

  float bnum = -1.0f;   // inter of current best
  float bden =  1.0f;   // union of current best
  int   bidx =  0;
  #pragma unroll 4
  for (int g = 0; g < NG; ++g) {
    float4 bb = sbox[g];                 // wave-uniform ds_load_b128 broadcast
    float iw   = fmaxf(fminf(ax2, bb.z) - fmaxf(ax1, bb.x), 0.0f);
    float ih   = fmaxf(fminf(ay2, bb.w) - fmaxf(ay1, bb.y), 0.0f);
    float inter = iw * ih;
    float uni   = areaA + sarea[g] - inter;
    // inter/uni > bnum/bden  <=>  inter*bden > bnum*uni   (uni,bden > 0)
    bool upd = inter * bden > bnum * uni;
    bnum = upd ? inter : bnum;
    bden = upd ? uni   : bden;
    bidx = upd ? g     : bidx;
  }

  const float maxov = bnum / fmaxf(bden, 1e-8f);
  const float state = (maxov >= 0.5f) ? 1.0f : ((maxov < 0.4f) ? 0.0f : -1.0f);

  const float4 gb  = sbox[bidx];
  const int    lab = slab[bidx];

  // bbox transform: ((gt-anchor)/size - mean)/std with mean=0, std=0.2 -> *5
  const float inv_wa = 5.0f / (ax2 - ax1);
  const float inv_ha = 5.0f / (ay2 - ay1);
  float4 rt;
  rt.x = (gb.x - ax1) * inv_wa;
  rt.y = (gb.y - ay1) * inv_ha;
  rt.z = (gb.z - ax2) * inv_wa;
  rt.w = (gb.w - ay2) * inv_ha;

  st[gid] = state;
  *(float4*)(reg + (size_t)gid * 4) = rt;

  const int code = (state == 1.0f) ? lab : -1;
  if (codes) codes[gid] = code;

  if (inline_cls) {  // fallback when workspace too small: strided cls writes
    float* cp = cls + (size_t)gid * NUM_CLASSES;
    #pragma unroll
    for (int c4 = 0; c4 < NUM_CLASSES / 4; ++c4) {
      const int c0 = c4 * 4;
      float4 v;
      v.x = (code == c0 + 0) ? 1.0f : 0.0f;
      v.y = (code == c0 + 1) ? 1.0f : 0.0f;
      v.z = (code == c0 + 2) ? 1.0f : 0.0f;
      v.w = (code == c0 + 3) ? 1.0f : 0.0f;
      *(float4*)(cp + c0) = v;
    }
  }
}

// Expand compact per-anchor class codes into the dense one-hot cls tensor
// with fully coalesced 128-bit stores (this is the 335 MB bandwidth hog).
__global__ __launch_bounds__(256) void cls_expand_kernel(
    const int* __restrict__ codes,   // (B*A)
    float*     __restrict__ cls)     // (B*A*80)
{
  const unsigned i = blockIdx.x * 256u + threadIdx.x;   // one float4 per thread
  const unsigned anchor = i / (NUM_CLASSES / 4);        // /20
  const int      c0     = (int)(i % (NUM_CLASSES / 4)) * 4;
  const int      code   = codes[anchor];
  float4 v;
  v.x = (code == c0 + 0) ? 1.0f : 0.0f;
  v.y = (code == c0 + 1) ? 1.0f : 0.0f;
  v.z = (code == c0 + 2) ? 1.0f : 0.0f;
  v.w = (code == c0 + 3) ? 1.0f : 0.0f;
  ((float4*)cls)[i] = v;
}

extern "C" void kernel_launch(void* const* d_in, const int* in_sizes, int n_in,
                              void* d_out, int out_size, void* d_ws, size_t ws_size,
                              hipStream_t stream) {
  const float* ann = (const float*)d_in[0];  // (8,128,5)
  const float* anc = (const float*)d_in[1];  // (131072,4)

  float* out = (float*)d_out;
  float* cls = out;                                    // B*A*80
  float* reg = out + (size_t)NB * NA * NUM_CLASSES;    // B*A*4
  float* st  = reg + (size_t)NB * NA * 4;              // B*A

  const bool two_phase = ws_size >= (size_t)NB * NA * sizeof(int);
  int* codes = two_phase ? (int*)d_ws : nullptr;

  anchor_assign_kernel<<<(NB * NA) / 256, 256, 0, stream>>>(
      ann, anc, cls, reg, st, codes, two_phase ? 0 : 1);

  if (two_phase) {
    const unsigned n4 = (unsigned)NB * NA * (NUM_CLASSES / 4);  // 20,971,520
    cls_expand_kernel<<<n4 / 256, 256, 0, stream>>>(codes, cls);
  }
}